// CustomMSDeformableAttention_39797166964900
// MI455X (gfx1250) — compile-verified
//
#include <hip/hip_runtime.h>
#include <hip/hip_bf16.h>

// ---------------- compile-time problem shape ----------------
#define NQ      2000
#define BSZ     8
#define EMBED   256
#define HEADS   8
#define POINTS  4
#define NV      20000      // 200*100
#define IMG_H   200
#define IMG_W   100
#define D_HEAD  32         // EMBED/HEADS

typedef __attribute__((ext_vector_type(2))) float v2f;
typedef __attribute__((ext_vector_type(8))) float v8f;

#if __has_builtin(__builtin_amdgcn_wmma_f32_16x16x4_f32)
#define USE_WMMA_F32 1
#else
#define USE_WMMA_F32 0
#endif

// =====================================================================
// Kernel 1: value projection  Vp[b][pos][e] = value[pos][b][:] @ W_val + b_val
// M = BSZ*NV = 160000 rows (r = b*NV + pos), N = 256, K = 256.
// One wave32 computes a 16x16 tile with V_WMMA_F32_16X16X4_F32, K in steps of 4.
// =====================================================================
__global__ void __launch_bounds__(128)
vproj_wmma_kernel(const float* __restrict__ value,
                  const float* __restrict__ Wv,
                  const float* __restrict__ bv,
                  float* __restrict__ Vp)
{
    const int wave = threadIdx.x >> 5;
    const int lane = threadIdx.x & 31;
    const int tile = blockIdx.x * 4 + wave;          // tiles = 10000 * 16
    const int tm = tile >> 4;                        // 0..9999  (M tile)
    const int tn = tile & 15;                        // 0..15    (N tile)
    const int row0 = tm * 16;
    const int col0 = tn * 16;
    const int half = lane >> 4;                      // 0 or 1
    const int l15  = lane & 15;

    // A-row for this lane: global row r -> value[(pos*BSZ + b)*EMBED]
    const int r   = row0 + l15;
    const int b   = r / NV;
    const int pos = r - b * NV;
    const float* __restrict__ arow = value + ((size_t)pos * BSZ + b) * EMBED;

    v8f acc = {};
#if USE_WMMA_F32
    const int n = col0 + l15;
    for (int k = 0; k < EMBED; k += 4) {
        const int kk = k + half * 2;
        v2f a, bb;
        a.x  = arow[kk];
        a.y  = arow[kk + 1];
        bb.x = Wv[(size_t)kk * EMBED + n];
        bb.y = Wv[(size_t)(kk + 1) * EMBED + n];
        acc = __builtin_amdgcn_wmma_f32_16x16x4_f32(
            false, a, false, bb, (short)0, acc, false, false);
    }
#else
    // scalar fallback producing the same D layout
    for (int i = 0; i < 8; ++i) {
        const int m  = row0 + i + 8 * half;
        const int bb2 = m / NV, pp = m - bb2 * NV;
        const float* ar = value + ((size_t)pp * BSZ + bb2) * EMBED;
        float s = 0.f;
        for (int k = 0; k < EMBED; ++k)
            s += ar[k] * Wv[(size_t)k * EMBED + col0 + l15];
        acc[i] = s;
    }
#endif
    const float bias = bv[col0 + l15];
    for (int i = 0; i < 8; ++i) {
        const int m = row0 + i + 8 * half;           // D: vgpr i holds M = i + 8*half
        Vp[(size_t)m * EMBED + col0 + l15] = acc[i] + bias;
    }
}

// =====================================================================
// Kernel 2: sampling offsets + attention-weight softmax per (b, q)
//   off_out: (BSZ*NQ, 64)   aw_out: (BSZ*NQ, 32)
// =====================================================================
__global__ void __launch_bounds__(128)
offattn_kernel(const float* __restrict__ query,
               const float* __restrict__ W_off, const float* __restrict__ b_off,
               const float* __restrict__ W_attn, const float* __restrict__ b_attn,
               float* __restrict__ off_out, float* __restrict__ aw_out)
{
    const int bq = blockIdx.x;              // b*NQ + q
    const int b  = bq / NQ;
    const int qi = bq - b * NQ;
    const float* __restrict__ qrow = query + ((size_t)qi * BSZ + b) * EMBED;

    __shared__ float qs[EMBED];
    __shared__ float logits[HEADS * POINTS];

    for (int i = threadIdx.x; i < EMBED; i += blockDim.x) qs[i] = qrow[i];
    __syncthreads();

    const int t = threadIdx.x;
    if (t < HEADS * POINTS * 2) {           // 64 offset outputs
        float s = b_off[t];
        for (int k = 0; k < EMBED; ++k) s += qs[k] * W_off[(size_t)k * 64 + t];
        off_out[(size_t)bq * 64 + t] = s;
    } else if (t < HEADS * POINTS * 2 + HEADS * POINTS) {   // 32 attn logits
        const int n = t - 64;
        float s = b_attn[n];
        for (int k = 0; k < EMBED; ++k) s += qs[k] * W_attn[(size_t)k * 32 + n];
        logits[n] = s;
    }
    __syncthreads();
    if (t < HEADS) {                        // softmax over 4 points per head
        float m = -1e30f;
        for (int p = 0; p < POINTS; ++p) m = fmaxf(m, logits[t * POINTS + p]);
        float e[POINTS], sum = 0.f;
        for (int p = 0; p < POINTS; ++p) { e[p] = __expf(logits[t * POINTS + p] - m); sum += e[p]; }
        const float inv = 1.0f / sum;
        for (int p = 0; p < POINTS; ++p)
            aw_out[(size_t)bq * 32 + t * POINTS + p] = e[p] * inv;
    }
}

// =====================================================================
// Kernel 3: bilinear gather + weighted point sum.
// One wave per (b, q, head); lane = channel d (0..31).
// Vp is 164 MB -> fits MI455X 192 MB L2, so random taps are L2-resident.
// =====================================================================
__device__ __forceinline__ float tap_read(const float* __restrict__ Vp,
                                          int b, int h, int x, int y, int d)
{
    if (x < 0 || x >= IMG_W || y < 0 || y >= IMG_H) return 0.f;
    return Vp[(((size_t)b * NV + (size_t)y * IMG_W + x) * EMBED) + h * D_HEAD + d];
}

__global__ void __launch_bounds__(256)
sample_kernel(const float* __restrict__ Vp,
              const float* __restrict__ refp,
              const float* __restrict__ off,
              const float* __restrict__ aw,
              float* __restrict__ out_pre)
{
    const int gwave = blockIdx.x * 8 + (threadIdx.x >> 5);   // 0 .. BSZ*NQ*HEADS-1
    const int lane  = threadIdx.x & 31;                      // channel d
    const int h  = gwave & (HEADS - 1);
    const int bq = gwave >> 3;                               // b*NQ + q

    const float rx = refp[(size_t)bq * 2 + 0];
    const float ry = refp[(size_t)bq * 2 + 1];
    const int b = bq / NQ;

    float acc = 0.f;
    for (int p = 0; p < POINTS; ++p) {
        const float ox = off[(size_t)bq * 64 + h * 8 + p * 2 + 0];
        const float oy = off[(size_t)bq * 64 + h * 8 + p * 2 + 1];
        // loc = ref + off/[W,H]; px = loc*W - 0.5, py = loc*H - 0.5
        const float px = (rx + ox * (1.0f / IMG_W)) * (float)IMG_W - 0.5f;
        const float py = (ry + oy * (1.0f / IMG_H)) * (float)IMG_H - 0.5f;
        const float x0f = floorf(px), y0f = floorf(py);
        const float fx = px - x0f, fy = py - y0f;
        const int x0 = (int)x0f, y0 = (int)y0f;

        const float v00 = tap_read(Vp, b, h, x0,     y0,     lane);
        const float v01 = tap_read(Vp, b, h, x0 + 1, y0,     lane);
        const float v10 = tap_read(Vp, b, h, x0,     y0 + 1, lane);
        const float v11 = tap_read(Vp, b, h, x0 + 1, y0 + 1, lane);

        const float s = v00 * (1.f - fx) * (1.f - fy) + v01 * fx * (1.f - fy)
                      + v10 * (1.f - fx) * fy         + v11 * fx * fy;
        acc += aw[(size_t)bq * 32 + h * POINTS + p] * s;
    }
    out_pre[(size_t)bq * EMBED + h * D_HEAD + lane] = acc;
}

// =====================================================================
// Kernel 4: output projection + bias + residual.
// M = BSZ*NQ = 16000 rows (r = b*NQ + q), N = K = 256.
// out[(q*BSZ+b)*EMBED + n] = out_pre[r] @ W_out + b_out[n] + query[(q*BSZ+b)*EMBED+n]
// =====================================================================
__global__ void __launch_bounds__(128)
outproj_wmma_kernel(const float* __restrict__ out_pre,
                    const float* __restrict__ Wo,
                    const float* __restrict__ bo,
                    const float* __restrict__ query,
                    float* __restrict__ out)
{
    const int wave = threadIdx.x >> 5;
    const int lane = threadIdx.x & 31;
    const int tile = blockIdx.x * 4 + wave;          // tiles = 1000 * 16
    const int tm = tile >> 4;
    const int tn = tile & 15;
    const int row0 = tm * 16;
    const int col0 = tn * 16;
    const int half = lane >> 4;
    const int l15  = lane & 15;

    const float* __restrict__ arow = out_pre + (size_t)(row0 + l15) * EMBED;

    v8f acc = {};
#if USE_WMMA_F32
    const int n = col0 + l15;
    for (int k = 0; k < EMBED; k += 4) {
        const int kk = k + half * 2;
        v2f a, bb;
        a.x  = arow[kk];
        a.y  = arow[kk + 1];
        bb.x = Wo[(size_t)kk * EMBED + n];
        bb.y = Wo[(size_t)(kk + 1) * EMBED + n];
        acc = __builtin_amdgcn_wmma_f32_16x16x4_f32(
            false, a, false, bb, (short)0, acc, false, false);
    }
#else
    for (int i = 0; i < 8; ++i) {
        const int m = row0 + i + 8 * half;
        const float* ar = out_pre + (size_t)m * EMBED;
        float s = 0.f;
        for (int k = 0; k < EMBED; ++k)
            s += ar[k] * Wo[(size_t)k * EMBED + col0 + l15];
        acc[i] = s;
    }
#endif
    const float bias = bo[col0 + l15];
    for (int i = 0; i < 8; ++i) {
        const int m  = row0 + i + 8 * half;          // r = b*NQ + q
        const int b  = m / NQ;
        const int qi = m - b * NQ;
        const size_t oidx = ((size_t)qi * BSZ + b) * EMBED + col0 + l15;
        out[oidx] = acc[i] + bias + query[oidx];
    }
}

// =====================================================================
extern "C" void kernel_launch(void* const* d_in, const int* in_sizes, int n_in,
                              void* d_out, int out_size, void* d_ws, size_t ws_size,
                              hipStream_t stream)
{
    const float* query  = (const float*)d_in[0];
    const float* value  = (const float*)d_in[1];
    const float* refp   = (const float*)d_in[2];
    const float* W_off  = (const float*)d_in[3];
    const float* b_off  = (const float*)d_in[4];
    const float* W_attn = (const float*)d_in[5];
    const float* b_attn = (const float*)d_in[6];
    const float* W_val  = (const float*)d_in[7];
    const float* b_val  = (const float*)d_in[8];
    const float* W_out  = (const float*)d_in[9];
    const float* b_out  = (const float*)d_in[10];
    float*       out    = (float*)d_out;

    // workspace partition
    char* ws = (char*)d_ws;
    float* Vp      = (float*)ws;                               // 160000*256 f32 = 164 MB
    ws += (size_t)BSZ * NV * EMBED * sizeof(float);
    float* out_pre = (float*)ws;                               // 16000*256 f32
    ws += (size_t)BSZ * NQ * EMBED * sizeof(float);
    float* off     = (float*)ws;                               // 16000*64 f32
    ws += (size_t)BSZ * NQ * 64 * sizeof(float);
    float* aw      = (float*)ws;                               // 16000*32 f32

    // K1: value projection (10000*16 tiles, 4 waves/block)
    vproj_wmma_kernel<<<(10000 * 16) / 4, 128, 0, stream>>>(value, W_val, b_val, Vp);
    // K2: offsets + attention softmax
    offattn_kernel<<<BSZ * NQ, 128, 0, stream>>>(query, W_off, b_off, W_attn, b_attn, off, aw);
    // K3: bilinear gather (BSZ*NQ*HEADS waves, 8 waves/block)
    sample_kernel<<<(BSZ * NQ * HEADS) / 8, 256, 0, stream>>>(Vp, refp, off, aw, out_pre);
    // K4: output projection + residual (1000*16 tiles, 4 waves/block)
    outproj_wmma_kernel<<<(1000 * 16) / 4, 128, 0, stream>>>(out_pre, W_out, b_out, query, out);
}